// AutoFlow_8847632630055
// MI455X (gfx1250) — compile-verified
//
#include <hip/hip_runtime.h>
#include <stdint.h>

// ---------------------------------------------------------------------------
// Embedding-row gather: out[i, :] = data[x[i], :]
//   x:    65536 int32 indices in [0, 100000)
//   data: 100000 x 512 fp32   (row = 2 KiB)
//   out:  65536 x 512 fp32
//
// HBM-bound (268 MB -> ~11.5 us @ 23.3 TB/s). Uses the CDNA5 Tensor Data
// Mover in GATHER mode: one D# descriptor DMAs 8 rows (32-bit index mode
// maximum) into LDS; the wave then streams the contiguous 16 KiB LDS block
// to the contiguous output slice with non-temporal b128 stores.
// ---------------------------------------------------------------------------

typedef unsigned int u32x4 __attribute__((ext_vector_type(4)));
typedef int          i32x4 __attribute__((ext_vector_type(4)));
typedef int          i32x8 __attribute__((ext_vector_type(8)));
typedef float        f32x4 __attribute__((ext_vector_type(4)));

#define OUT_DIM        512
#define NB_DATA_ROWS   100000u
#define GROUP          8                      // rows per TDM gather (32-bit idx max)
#define WAVES_PER_BLK  4
#define BLOCK_THREADS  (WAVES_PER_BLK * 32)
#define GROUP_FLOATS   (GROUP * OUT_DIM)      // 4096 floats = 16 KiB
#define GROUP_VEC4     (GROUP_FLOATS / 4)     // 1024 float4
#define VEC_ITERS      (GROUP_VEC4 / 32)      // 32 iterations per wave

__global__ __launch_bounds__(BLOCK_THREADS)
void gather_rows_tdm(const int* __restrict__ idx,
                     const float* __restrict__ data,
                     float* __restrict__ out,
                     int n_rows)
{
    __shared__ float smem[WAVES_PER_BLK * GROUP_FLOATS];   // 64 KiB

    // readfirstlane makes the wave id compiler-visibly uniform: descriptor
    // building stays in SALU and index fetches scalarize.
    const int wave  = __builtin_amdgcn_readfirstlane((int)(threadIdx.x >> 5));
    const int lane  = (int)(threadIdx.x & 31);
    const int group = (int)blockIdx.x * WAVES_PER_BLK + wave;  // group of 8 rows
    const long row0 = (long)group * GROUP;
    if (row0 >= (long)n_rows) return;
    const long rem       = (long)n_rows - row0;
    const int  rows_here = rem < (long)GROUP ? (int)rem : GROUP;

    float* lds = &smem[wave * GROUP_FLOATS];
    const uint32_t lds_addr = (uint32_t)(uintptr_t)(void*)lds; // low 32 = LDS byte offset

    // The 8 row indices for this group (uniform address -> scalar loads).
    const int* gi = idx + row0;
    uint32_t r[GROUP];
    if (rows_here == GROUP) {
#pragma unroll
        for (int j = 0; j < GROUP; ++j) r[j] = (uint32_t)gi[j];
    } else {
#pragma unroll
        for (int j = 0; j < GROUP; ++j)
            r[j] = (j < rows_here) ? (uint32_t)gi[j] : 0u;
    }

    const uint64_t gaddr = (uint64_t)(uintptr_t)data;

    // ---- Tensor DMA descriptor (D#), gather mode, 32-bit row indices ----
    // Group 0 (4 SGPRs)
    u32x4 g0;
    g0[0] = 1u                                  // count = 1 (valid descriptor)
          | (1u << 30)                          // gather_index_size = 32-bit
          | (1u << 31);                         // gather_mode = 1
    g0[1] = lds_addr;                           // lds_addr (bytes)
    g0[2] = (uint32_t)(gaddr & 0xFFFFFFFFull);  // global_addr[31:0]
    g0[3] = (uint32_t)((gaddr >> 32) & 0x1FFFFFFull)   // global_addr[56:32]
          | (2u << 30);                         // type = 2 ("image")

    // Group 1 (8 SGPRs)
    i32x8 g1;
    g1[0] = (int)(2u << 16);                            // workgroup_mask=0, data_size=2 (4B)
    g1[1] = (int)(((uint32_t)OUT_DIM & 0xFFFFu) << 16); // tensor_dim0[15:0]  @ bits 63:48
    g1[2] = (int)((((uint32_t)OUT_DIM >> 16) & 0xFFFFu) // tensor_dim0[31:16]
          | ((NB_DATA_ROWS & 0xFFFFu) << 16));          // tensor_dim1[15:0]
    g1[3] = (int)(((NB_DATA_ROWS >> 16) & 0xFFFFu)      // tensor_dim1[31:16]
          | (((uint32_t)OUT_DIM & 0xFFFFu) << 16));     // tile_dim0 = 512
    g1[4] = rows_here;                                  // tile_dim1 = #valid indices
    g1[5] = OUT_DIM;                                    // tensor_dim0_stride[31:0] = 512
    g1[6] = 0;                                          // stride hi / dim1_stride (ignored)
    g1[7] = 0;

    // Groups 2 & 3: row indices (32-bit mode -> 4 per group)
    i32x4 g2 = { (int)r[0], (int)r[1], (int)r[2], (int)r[3] };
    i32x4 g3 = { (int)r[4], (int)r[5], (int)r[6], (int)r[7] };
    i32x8 gpad = { 0, 0, 0, 0, 0, 0, 0, 0 };            // unused trailing group

    // Issue the gather DMA: memory -> LDS (wave-scalar; EXEC ignored).
    // amdgpu-toolchain (clang-23) 6-arg form: (g0, g1, g2, g3, pad, cpol)
    __builtin_amdgcn_tensor_load_to_lds(g0, g1, g2, g3, gpad, 0);

    // Wait for the tensor op to land in LDS.
    __builtin_amdgcn_s_wait_tensorcnt(0);

    // The 8 gathered rows are contiguous in LDS and contiguous in the output:
    // stream 16 KiB out with non-temporal 128-bit stores (protect L2 for the
    // 200 MB gather table; output is never re-read).
    const f32x4* src = (const f32x4*)lds;
    f32x4*       dst = (f32x4*)(out + row0 * OUT_DIM);
    if (rows_here == GROUP) {
#pragma unroll
        for (int i = 0; i < VEC_ITERS; ++i) {
            f32x4 v = src[i * 32 + lane];
            __builtin_nontemporal_store(v, &dst[i * 32 + lane]);
        }
    } else {
        const int nvec = rows_here * (OUT_DIM / 4);
        for (int i = lane; i < nvec; i += 32) {
            f32x4 v = src[i];
            __builtin_nontemporal_store(v, &dst[i]);
        }
    }
}

extern "C" void kernel_launch(void* const* d_in, const int* in_sizes, int n_in,
                              void* d_out, int out_size, void* d_ws, size_t ws_size,
                              hipStream_t stream) {
    const int*   x    = (const int*)d_in[0];    // 16384*4*1 = 65536 indices
    const float* data = (const float*)d_in[1];  // 100000 x 512 fp32
    float*       out  = (float*)d_out;          // 65536 x 512 fp32

    const int n_rows = in_sizes[0];             // flat count of x = gathered rows
    const int groups = (n_rows + GROUP - 1) / GROUP;
    const int blocks = (groups + WAVES_PER_BLK - 1) / WAVES_PER_BLK;

    gather_rows_tdm<<<blocks, BLOCK_THREADS, 0, stream>>>(x, data, out, n_rows);
}